// GAT_3384434229555
// MI455X (gfx1250) — compile-verified
//
#include <hip/hip_runtime.h>
#include <math.h>

// ---------------------------------------------------------------------------
// Two-layer GATConv for MI455X (gfx1250, wave32).
// GEMMs use native fp32 WMMA: V_WMMA_F32_16X16X4_F32 (one wave per 16x16 tile).
// Edge softmax: uint-keyed atomic max -> atomic-add denom -> atomic scatter.
// Whole working set (~180 MB) is L2-resident (192 MB), so edge passes are
// L2-bound; per-node attention arrays are recomputed instead of materializing
// per-edge tensors.
// ---------------------------------------------------------------------------

typedef float v2f __attribute__((ext_vector_type(2)));
typedef float v8f __attribute__((ext_vector_type(8)));

#define IN_C      128
#define HID       16
#define HEADS     8
#define OUT_C     64
#define NEG_SLOPE 0.2f
#define GAT_EPS   1e-16f

__device__ __forceinline__ unsigned f2key(float f) {
  unsigned u = __float_as_uint(f);
  return (u & 0x80000000u) ? ~u : (u | 0x80000000u);
}
__device__ __forceinline__ float key2f(unsigned u) {
  return (u & 0x80000000u) ? __uint_as_float(u & 0x7FFFFFFFu)
                           : __uint_as_float(~u);
}
__device__ __forceinline__ float lrelu(float x) { return x > 0.f ? x : NEG_SLOPE * x; }
__device__ __forceinline__ float eluf(float x)  { return x > 0.f ? x : (__expf(x) - 1.f); }

// ---------------------------------------------------------------------------
// C[M,NC] = A[M,K] * B[K,NC], row-major, all dims multiples of 16, K mult of 4.
// One wave (32 threads) per 16x16 tile of C. grid = (M/16, NC/16).
// A-frag (16x4 f32): lane l<16 holds row l, K={k0,k0+1}; lane l+16 holds K={k0+2,k0+3}.
// B-frag (4x16 f32): VGPR0 = row k0 (lanes 0-15) / row k0+2 (lanes 16-31), VGPR1 = +1.
// D: VGPR i, lanes 0-15 -> M=i, lanes 16-31 -> M=i+8; N = lane&15.
// ---------------------------------------------------------------------------
__global__ void k_wmma_gemm_f32(const float* __restrict__ A,
                                const float* __restrict__ B,
                                float* __restrict__ C, int K, int NC) {
  const int lane = threadIdx.x;        // 0..31 (wave32)
  const int half = lane >> 4;          // 0 or 1
  const int l    = lane & 15;
  const int row0 = blockIdx.x * 16;
  const int col0 = blockIdx.y * 16;

  const float* arow = A + (size_t)(row0 + l) * K + 2 * half;   // 8B-aligned
  const float* bcol = B + (size_t)(2 * half) * NC + col0 + l;

  v8f acc = {0.f, 0.f, 0.f, 0.f, 0.f, 0.f, 0.f, 0.f};
  for (int k0 = 0; k0 < K; k0 += 4) {
    v2f a = *(const v2f*)(arow + k0);
    v2f b;
    b.x = bcol[(size_t)k0 * NC];
    b.y = bcol[(size_t)(k0 + 1) * NC];
    acc = __builtin_amdgcn_wmma_f32_16x16x4_f32(false, a, false, b,
                                                (short)0, acc, false, false);
  }

  float* crow = C + (size_t)(row0 + 8 * half) * NC + col0 + l;
#pragma unroll
  for (int i = 0; i < 8; ++i) crow[(size_t)i * NC] = acc[i];
}

// ---------------- layer 1 helpers ----------------

// out1[n,c] = b1[c]; m1 = 0-key (== -inf); den1 = 0
__global__ void k_init1(float* __restrict__ out1, unsigned* __restrict__ m1,
                        float* __restrict__ den1, const float* __restrict__ b1,
                        int n) {
  int t = blockIdx.x * blockDim.x + threadIdx.x;
  if (t < n * (HEADS * HID)) out1[t] = b1[t & (HEADS * HID - 1)];
  if (t < n * HEADS) { m1[t] = 0u; den1[t] = 0.f; }
}

// asrc1[n,h] = sum_c h1[n,h,c]*a_src1[h,c]  (and adst1 likewise)
__global__ void k_att1(const float* __restrict__ h1,
                       const float* __restrict__ a_src,
                       const float* __restrict__ a_dst,
                       float* __restrict__ asrc, float* __restrict__ adst, int n) {
  int t = blockIdx.x * blockDim.x + threadIdx.x;
  if (t >= n * HEADS) return;
  const int hd = t & (HEADS - 1);
  const float* hp = h1 + (size_t)(t >> 3) * (HEADS * HID) + hd * HID;
  const float* as = a_src + hd * HID;
  const float* ad = a_dst + hd * HID;
  float s = 0.f, d = 0.f;
#pragma unroll
  for (int c = 0; c < HID; ++c) { float v = hp[c]; s += v * as[c]; d += v * ad[c]; }
  asrc[t] = s;
  adst[t] = d;
}

__device__ __forceinline__ void edge_sd(const int* __restrict__ ei, int E, int n,
                                        int e, int& s, int& d) {
  if (e < E) { s = ei[e]; d = ei[E + e]; } else { s = d = e - E; }
}

// segment max (uint-keyed) of leakyrelu(asrc[s]+adst[d]) into m1[d,h]
__global__ void k_max1(const int* __restrict__ ei, int E, int n,
                       const float* __restrict__ asrc, const float* __restrict__ adst,
                       unsigned* __restrict__ m1) {
  int e = blockIdx.x * blockDim.x + threadIdx.x;
  if (e >= E + n) return;
  int s, d; edge_sd(ei, E, n, e, s, d);
  const float* as = asrc + (size_t)s * HEADS;
  const float* ad = adst + (size_t)d * HEADS;
  unsigned* mp = m1 + (size_t)d * HEADS;
#pragma unroll
  for (int h = 0; h < HEADS; ++h)
    atomicMax(mp + h, f2key(lrelu(as[h] + ad[h])));
}

// den1[d,h] += exp(e - m[d,h])
__global__ void k_den1(const int* __restrict__ ei, int E, int n,
                       const float* __restrict__ asrc, const float* __restrict__ adst,
                       const unsigned* __restrict__ m1, float* __restrict__ den1) {
  int e = blockIdx.x * blockDim.x + threadIdx.x;
  if (e >= E + n) return;
  int s, d; edge_sd(ei, E, n, e, s, d);
  const float* as = asrc + (size_t)s * HEADS;
  const float* ad = adst + (size_t)d * HEADS;
  const unsigned* mp = m1 + (size_t)d * HEADS;
  float* dp = den1 + (size_t)d * HEADS;
#pragma unroll
  for (int h = 0; h < HEADS; ++h) {
    float v = lrelu(as[h] + ad[h]);
    atomicAdd(dp + h, __expf(v - key2f(mp[h])));
  }
}

// out1[d, h, :] += alpha * h1[s, h, :]   one thread per (edge, head)
__global__ void k_scat1(const int* __restrict__ ei, int E, int n,
                        const float* __restrict__ asrc, const float* __restrict__ adst,
                        const unsigned* __restrict__ m1, const float* __restrict__ den1,
                        const float* __restrict__ h1, float* __restrict__ out1) {
  int t = blockIdx.x * blockDim.x + threadIdx.x;
  if (t >= (E + n) * HEADS) return;
  const int e = t >> 3, h = t & (HEADS - 1);
  int s, d; edge_sd(ei, E, n, e, s, d);
  const float v = lrelu(asrc[(size_t)s * HEADS + h] + adst[(size_t)d * HEADS + h]);
  const float alpha = __expf(v - key2f(m1[(size_t)d * HEADS + h])) /
                      (den1[(size_t)d * HEADS + h] + GAT_EPS);
  const float* hp = h1 + (size_t)s * (HEADS * HID) + h * HID;
  float* op = out1 + (size_t)d * (HEADS * HID) + h * HID;
#pragma unroll
  for (int c = 0; c < HID; ++c) atomicAdd(op + c, alpha * hp[c]);
}

__global__ void k_elu(float* __restrict__ out1, int n) {
  int t = blockIdx.x * blockDim.x + threadIdx.x;
  if (t < n * (HEADS * HID)) out1[t] = eluf(out1[t]);
}

// ---------------- layer 2 helpers (1 head, 64 channels) ----------------

__global__ void k_att2(const float* __restrict__ G,
                       const float* __restrict__ a_src2, const float* __restrict__ a_dst2,
                       float* __restrict__ asrc2, float* __restrict__ adst2, int n) {
  int t = blockIdx.x * blockDim.x + threadIdx.x;
  if (t >= n) return;
  const float* g = G + (size_t)t * OUT_C;
  float s = 0.f, d = 0.f;
#pragma unroll
  for (int c = 0; c < OUT_C; ++c) { float v = g[c]; s += v * a_src2[c]; d += v * a_dst2[c]; }
  asrc2[t] = s;
  adst2[t] = d;
}

__global__ void k_init2(float* __restrict__ out, unsigned* __restrict__ m2,
                        float* __restrict__ den2, const float* __restrict__ b2, int n) {
  int t = blockIdx.x * blockDim.x + threadIdx.x;
  if (t < n * OUT_C) out[t] = b2[t & (OUT_C - 1)];
  if (t < n) { m2[t] = 0u; den2[t] = 0.f; }
}

__global__ void k_max2(const int* __restrict__ ei, int E, int n,
                       const float* __restrict__ asrc2, const float* __restrict__ adst2,
                       unsigned* __restrict__ m2) {
  int e = blockIdx.x * blockDim.x + threadIdx.x;
  if (e >= E + n) return;
  int s, d; edge_sd(ei, E, n, e, s, d);
  atomicMax(m2 + d, f2key(lrelu(asrc2[s] + adst2[d])));
}

__global__ void k_den2(const int* __restrict__ ei, int E, int n,
                       const float* __restrict__ asrc2, const float* __restrict__ adst2,
                       const unsigned* __restrict__ m2, float* __restrict__ den2) {
  int e = blockIdx.x * blockDim.x + threadIdx.x;
  if (e >= E + n) return;
  int s, d; edge_sd(ei, E, n, e, s, d);
  float v = lrelu(asrc2[s] + adst2[d]);
  atomicAdd(den2 + d, __expf(v - key2f(m2[d])));
}

// one thread per (edge, 16-channel quarter)
__global__ void k_scat2(const int* __restrict__ ei, int E, int n,
                        const float* __restrict__ asrc2, const float* __restrict__ adst2,
                        const unsigned* __restrict__ m2, const float* __restrict__ den2,
                        const float* __restrict__ G, float* __restrict__ out) {
  int t = blockIdx.x * blockDim.x + threadIdx.x;
  if (t >= (E + n) * 4) return;
  const int e = t >> 2, q = t & 3;
  int s, d; edge_sd(ei, E, n, e, s, d);
  const float v = lrelu(asrc2[s] + adst2[d]);
  const float alpha = __expf(v - key2f(m2[d])) / (den2[d] + GAT_EPS);
  const float* g = G + (size_t)s * OUT_C + q * 16;
  float* op = out + (size_t)d * OUT_C + q * 16;
#pragma unroll
  for (int c = 0; c < 16; ++c) atomicAdd(op + c, alpha * g[c]);
}

// ---------------------------------------------------------------------------

static inline int cdiv(long long a, int b) { return (int)((a + b - 1) / b); }

extern "C" void kernel_launch(void* const* d_in, const int* in_sizes, int n_in,
                              void* d_out, int out_size, void* d_ws, size_t ws_size,
                              hipStream_t stream) {
  const float* x      = (const float*)d_in[0];
  const int*   ei     = (const int*)d_in[1];
  const float* W1     = (const float*)d_in[2];
  const float* a_src1 = (const float*)d_in[3];
  const float* a_dst1 = (const float*)d_in[4];
  const float* b1     = (const float*)d_in[5];
  const float* W2     = (const float*)d_in[6];
  const float* a_src2 = (const float*)d_in[7];
  const float* a_dst2 = (const float*)d_in[8];
  const float* b2     = (const float*)d_in[9];

  const int n  = in_sizes[0] / IN_C;  // 100000 (multiple of 16)
  const int E  = in_sizes[1] / 2;     // 1600000
  const int Et = E + n;               // with self-loops

  // workspace carve-up (fp32 unless noted)
  float*    h1    = (float*)d_ws;                         // n*128
  float*    out1  = h1 + (size_t)n * 128;                 // n*128 (becomes x2 after ELU)
  float*    G     = out1 + (size_t)n * 128;               // n*64
  float*    asrc1 = G + (size_t)n * 64;                   // n*8
  float*    adst1 = asrc1 + (size_t)n * HEADS;            // n*8
  float*    den1  = adst1 + (size_t)n * HEADS;            // n*8
  unsigned* m1    = (unsigned*)(den1 + (size_t)n * HEADS);// n*8
  float*    asrc2 = (float*)(m1 + (size_t)n * HEADS);     // n
  float*    adst2 = asrc2 + n;                            // n
  float*    den2  = adst2 + n;                            // n
  unsigned* m2    = (unsigned*)(den2 + n);                // n
  float*    out   = (float*)d_out;                        // n*64

  const int B = 256;

  // ---- layer 1 ----
  k_init1<<<cdiv((long long)n * 128, B), B, 0, stream>>>(out1, m1, den1, b1, n);
  k_wmma_gemm_f32<<<dim3(n / 16, IN_C / 16), 32, 0, stream>>>(x, W1, h1, IN_C, HEADS * HID);
  k_att1<<<cdiv((long long)n * HEADS, B), B, 0, stream>>>(h1, a_src1, a_dst1, asrc1, adst1, n);
  k_max1<<<cdiv(Et, B), B, 0, stream>>>(ei, E, n, asrc1, adst1, m1);
  k_den1<<<cdiv(Et, B), B, 0, stream>>>(ei, E, n, asrc1, adst1, m1, den1);
  k_scat1<<<cdiv((long long)Et * HEADS, B), B, 0, stream>>>(ei, E, n, asrc1, adst1, m1, den1, h1, out1);
  k_elu<<<cdiv((long long)n * 128, B), B, 0, stream>>>(out1, n);

  // ---- layer 2 ----
  k_wmma_gemm_f32<<<dim3(n / 16, OUT_C / 16), 32, 0, stream>>>(out1, W2, G, HEADS * HID, OUT_C);
  k_att2<<<cdiv(n, B), B, 0, stream>>>(G, a_src2, a_dst2, asrc2, adst2, n);
  k_init2<<<cdiv((long long)n * OUT_C, B), B, 0, stream>>>(out, m2, den2, b2, n);
  k_max2<<<cdiv(Et, B), B, 0, stream>>>(ei, E, n, asrc2, adst2, m2);
  k_den2<<<cdiv(Et, B), B, 0, stream>>>(ei, E, n, asrc2, adst2, m2, den2);
  k_scat2<<<cdiv((long long)Et * 4, B), B, 0, stream>>>(ei, E, n, asrc2, adst2, m2, den2, G, out);
}